// GCNConv_18476949307690
// MI455X (gfx1250) — compile-verified
//
#include <hip/hip_runtime.h>

// GCNConv on MI455X (gfx1250, wave32):
//   Phase 1: XW = X @ W via V_WMMA_F32_16X16X4_F32 (full fp32, matches reference numerics).
//            One wave computes a 16x64 strip: 4 accumulator tiles share each A-fragment,
//            so X (25.6 MB, the HBM-bound operand) is streamed exactly once.
//   Phase 2: out[i] = deg[i] * sum_{j in N(i)} deg[src_j] * XW[src_j]   (fixed degree 16).
//            XW lives in d_ws (25.6 MB) and fits in the 192 MB L2, so the nominal
//            410 MB of edge gathers are L2 hits; HBM traffic ~83 MB total.

#define NNODES 100000
#define DEG    16
#define DIM    64
#define MTILES (NNODES / 16)   // 6250 strips of 16 rows

typedef __attribute__((ext_vector_type(2))) float v2f;
typedef __attribute__((ext_vector_type(8))) float v8f;

#define WMMA_F32(A, B, C)                                               \
  __builtin_amdgcn_wmma_f32_16x16x4_f32(false, (A), false, (B),         \
                                        (short)0, (C), false, false)

// ---------------- Phase 1: dense GEMM with f32 WMMA ----------------
// A (16x4 f32) per-lane layout: lanes 0-15 -> M=lane, VGPR0=K0, VGPR1=K1;
//                               lanes 16-31 -> M=lane-16, VGPR0=K2, VGPR1=K3.
// B (4x16 f32) mirrored: lanes 0-15 -> N=lane, VGPR0=K0 row, VGPR1=K1 row;
//                        lanes 16-31 -> N=lane-16, VGPR0=K2 row, VGPR1=K3 row.
// C/D (16x16 f32): VGPR r holds M=r (lanes 0-15) and M=r+8 (lanes 16-31), N=lane%16.
__global__ __launch_bounds__(256) void gcn_gemm_wmma(
    const float* __restrict__ X, const float* __restrict__ W,
    float* __restrict__ XW) {
  const int gwave = (blockIdx.x * 256 + threadIdx.x) >> 5;  // one wave = one 16-row strip
  const int lane  = threadIdx.x & 31;
  if (gwave >= MTILES) return;                              // wave-uniform guard

  const int m0   = gwave * 16;
  const int half = lane >> 4;                               // 0: K0/K1, 1: K2/K3
  const int l    = lane & 15;

  const float* __restrict__ arow  = X + (size_t)(m0 + l) * DIM + 2 * half;
  const float* __restrict__ bbase = W + (size_t)(2 * half) * DIM + l;

  v8f c0 = {}, c1 = {}, c2 = {}, c3 = {};
#pragma unroll
  for (int k = 0; k < DIM; k += 4) {
    v2f a;                                   // one A-fragment feeds 4 WMMAs
    a.x = arow[k];
    a.y = arow[k + 1];
    const float* __restrict__ bk = bbase + (size_t)k * DIM;
    v2f b0, b1, b2, b3;
    b0.x = bk[0];  b0.y = bk[DIM + 0];       // W row k+2h / k+1+2h, cols n0+l
    b1.x = bk[16]; b1.y = bk[DIM + 16];
    b2.x = bk[32]; b2.y = bk[DIM + 32];
    b3.x = bk[48]; b3.y = bk[DIM + 48];
    c0 = WMMA_F32(a, b0, c0);
    c1 = WMMA_F32(a, b1, c1);
    c2 = WMMA_F32(a, b2, c2);
    c3 = WMMA_F32(a, b3, c3);
  }

  float* __restrict__ orow = XW + (size_t)m0 * DIM + l;
#pragma unroll
  for (int r = 0; r < 8; ++r) {
    const size_t ro = (size_t)(r + 8 * half) * DIM;
    orow[ro +  0] = c0[r];
    orow[ro + 16] = c1[r];
    orow[ro + 32] = c2[r];
    orow[ro + 48] = c3[r];
  }
}

// ---------------- Phase 2: edge aggregation ----------------
// One wave per node: lane L owns output features {2L, 2L+1}. The 16 neighbor
// indices are loaded once and broadcast via readlane so the XW row address is
// SGPR-based; each edge reads one contiguous 256 B row with a fully coalesced
// 32-lane float2 load (L2-resident).
__global__ __launch_bounds__(256) void gcn_aggregate(
    const float* __restrict__ XW, const int* __restrict__ col,
    const float* __restrict__ deg, float* __restrict__ out) {
  const int node = (blockIdx.x * 256 + threadIdx.x) >> 5;
  const int lane = threadIdx.x & 31;
  if (node >= NNODES) return;

  const int myidx = col[(size_t)node * DEG + (lane & 15)];
  const float di  = deg[node];

  float accx = 0.0f, accy = 0.0f;
#pragma unroll
  for (int j = 0; j < DEG; ++j) {
    const int s = __builtin_amdgcn_readlane(myidx, j);       // SGPR broadcast
    const float nrm = deg[s];                                // uniform load
    const float2 v = *(const float2*)(XW + (size_t)s * DIM + 2 * lane);
    accx = fmaf(nrm, v.x, accx);
    accy = fmaf(nrm, v.y, accy);
  }

  float2 r;
  r.x = di * accx;
  r.y = di * accy;
  *(float2*)(out + (size_t)node * DIM + 2 * lane) = r;
}

extern "C" void kernel_launch(void* const* d_in, const int* in_sizes, int n_in,
                              void* d_out, int out_size, void* d_ws, size_t ws_size,
                              hipStream_t stream) {
  const float* X   = (const float*)d_in[0];   // [100000, 64]
  const float* W   = (const float*)d_in[1];   // [64, 64]
  // d_in[2] = row_pointers: unused (degree fixed at 16 by construction)
  const int*   col = (const int*)d_in[3];     // [1600000]
  const float* deg = (const float*)d_in[4];   // [100000]
  float* out = (float*)d_out;                 // [100000, 64]
  float* xw  = (float*)d_ws;                  // scratch: 100000*64*4 = 25.6 MB

  // 6250 strip-waves -> 782 blocks of 8 waves (guard is wave-uniform; EXEC all-1s for WMMA)
  gcn_gemm_wmma<<<(MTILES + 7) / 8, 256, 0, stream>>>(X, W, xw);
  // 100000 node-waves -> 12500 blocks of 8 waves
  gcn_aggregate<<<12500, 256, 0, stream>>>(xw, col, deg, out);
}